// QuantumLayer_62466004353753
// MI455X (gfx1250) — compile-verified
//
#include <hip/hip_runtime.h>
#include <hip/hip_bf16.h>

typedef float v2f __attribute__((ext_vector_type(2)));
typedef float v4f __attribute__((ext_vector_type(4)));
typedef float v8f __attribute__((ext_vector_type(8)));

#define WMMA_F32(A, B, C) \
  __builtin_amdgcn_wmma_f32_16x16x4_f32(false, (A), false, (B), (short)0, (C), false, false)

__device__ __forceinline__ float2 cmul(float2 a, float2 b) {
  return make_float2(a.x * b.x - a.y * b.y, a.x * b.y + a.y * b.x);
}

// ---------------------------------------------------------------------------
// Kernel 1: build V = U(p)^T (row t of V = parameter circuit applied to e_t),
// stored k-pair interleaved for b128 B-fragment loads:
//   W[((k>>1)*256 + n)*4 + {0,1,2,3}] = {Vr[k][n], Vr[k+1][n], Vi[k][n], Vi[k+1][n]}
// One block (256 threads) per row; state in LDS (2KB).  Negligible cost.
// ---------------------------------------------------------------------------
struct CT { int c, t; };

__device__ __forceinline__ void gate2x2(float2* st, int s,
                                        float2 u00, float2 u01,
                                        float2 u10, float2 u11, int wire) {
  int mask = 1 << (7 - wire);
  float2 my = st[s];
  float2 ot = st[s ^ mask];
  float2 nv;
  if (s & mask) {
    float2 t0 = cmul(u10, ot), t1 = cmul(u11, my);
    nv = make_float2(t0.x + t1.x, t0.y + t1.y);
  } else {
    float2 t0 = cmul(u00, my), t1 = cmul(u01, ot);
    nv = make_float2(t0.x + t1.x, t0.y + t1.y);
  }
  __syncthreads();
  st[s] = nv;
  __syncthreads();
}

// kind: 0 = RY, 1 = RX, 2 = RZ.  Gate angle is 2*p -> half angle h = p.
__device__ __forceinline__ void rot_layer(float2* st, int s,
                                          const float* __restrict__ p,
                                          int base, int kind) {
  for (int i = 0; i < 8; ++i) {
    float h = p[base + i];
    float c = cosf(h), sn = sinf(h);
    float2 u00, u01, u10, u11;
    if (kind == 0) {         // RY: [[c,-s],[s,c]]
      u00 = make_float2(c, 0.f);  u01 = make_float2(-sn, 0.f);
      u10 = make_float2(sn, 0.f); u11 = make_float2(c, 0.f);
    } else if (kind == 1) {  // RX: [[c,-is],[-is,c]]
      u00 = make_float2(c, 0.f);   u01 = make_float2(0.f, -sn);
      u10 = make_float2(0.f, -sn); u11 = make_float2(c, 0.f);
    } else {                 // RZ: diag(e^{-ih}, e^{ih})
      u00 = make_float2(c, -sn);   u01 = make_float2(0.f, 0.f);
      u10 = make_float2(0.f, 0.f); u11 = make_float2(c, sn);
    }
    gate2x2(st, s, u00, u01, u10, u11, i);
  }
}

__device__ __forceinline__ void cnot_layer(float2* st, int s,
                                           const CT* L, int n) {
  for (int k = 0; k < n; ++k) {
    int cb = 1 << (7 - L[k].c);
    int tb = 1 << (7 - L[k].t);
    float2 v = (s & cb) ? st[s ^ tb] : st[s];
    __syncthreads();
    st[s] = v;
    __syncthreads();
  }
}

__global__ __launch_bounds__(256) void build_unitary(
    const float* __restrict__ p, float* __restrict__ W) {
  __shared__ float2 st[256];
  const int s = threadIdx.x;
  const int j = blockIdx.x;   // row of V (= k index of GEMM)
  st[s] = make_float2(s == j ? 1.f : 0.f, 0.f);
  __syncthreads();

  const CT L1[]  = {{0,1},{1,2},{2,3},{3,4},{4,5},{5,6},{6,7},{7,0}};
  const CT L2[]  = {{0,1},{2,3},{4,5},{6,7}};
  const CT L3[]  = {{1,3},{3,5},{5,7}};
  const CT L4[]  = {{0,4},{1,5},{2,6},{3,7}};
  const CT L5[]  = {{7,6},{6,5},{5,4},{4,3},{3,2},{2,1},{1,0}};
  const CT L6[]  = {{0,2},{2,4},{4,6}};
  const CT L7[]  = {{0,4},{1,5},{2,6},{3,7}};
  const CT L8[]  = {{0,3},{1,4},{2,5},{3,6},{4,7},{5,0},{6,1},{7,2}};
  const CT L9[]  = {{0,1},{0,2},{0,3},{0,4},{0,5},{0,6},{0,7}};
  const CT L10[] = {{0,1},{1,2},{2,3},{3,4},{4,5},{5,6},{6,7},{7,0}};

  rot_layer(st, s, p, 0, 0);   cnot_layer(st, s, L1, 8);
  rot_layer(st, s, p, 8, 1);   cnot_layer(st, s, L2, 4);
                               cnot_layer(st, s, L3, 3);
  rot_layer(st, s, p, 16, 2);  cnot_layer(st, s, L4, 4);
  rot_layer(st, s, p, 24, 0);  cnot_layer(st, s, L5, 7);
  rot_layer(st, s, p, 32, 1);  cnot_layer(st, s, L6, 3);
  rot_layer(st, s, p, 40, 2);  cnot_layer(st, s, L7, 4);
  rot_layer(st, s, p, 48, 0);  cnot_layer(st, s, L8, 8);
  rot_layer(st, s, p, 56, 1);  cnot_layer(st, s, L9, 7);
  rot_layer(st, s, p, 64, 2);  cnot_layer(st, s, L10, 8);
  rot_layer(st, s, p, 72, 0);

  // k-pair interleaved store: this row j contributes slot (j&1) of pair j>>1.
  const int base = ((j >> 1) * 256 + s) * 4;
  W[base + (j & 1)]     = st[s].x;   // re
  W[base + 2 + (j & 1)] = st[s].y;   // im
}

// ---------------------------------------------------------------------------
// Kernel 2: per block (128 threads = 4 waves) handle 64 batch rows.
// Phase 1: psi_init (product state) for the rows, k-pair interleaved in LDS
//          (row stride 516 floats: rows advance 4 banks -> conflict-free
//          ds_read_b128 A-fragments).
// Phase 2: complex GEMM psi_init (64x256) x V (256x256) with
//          v_wmma_f32_16x16x4_f32 (4 accumulators: RR, II, RI, IR), fused
//          sign-weighted |.|^2 reduction and cross-lane shfl_xor fold.
// Inner k-step: 1 ds_read_b128 + 1 global_load_b128 + 4 WMMA.
// ---------------------------------------------------------------------------
#define SROW 516
#define LDS_BYTES (64u * SROW * 4u)

__global__ __launch_bounds__(128) void qnn_gemm_reduce(
    const float* __restrict__ x, const float* __restrict__ W,
    float* __restrict__ out) {
  extern __shared__ float sP[];        // 64 rows x SROW floats

  const int tid  = threadIdx.x;
  const int wv   = tid >> 5;
  const int lane = tid & 31;
  const int half = lane >> 4;          // selects K pair within fragment
  const int lm   = lane & 15;          // row of M-tile / col of N-tile
  const int b0   = blockIdx.x * 64;
  const float inv_s2 = 0.70710678118654752f;

  // ---- Phase 1: psi_init for this wave's 16 rows ----
  for (int lr = 0; lr < 16; ++lr) {
    const int row = wv * 16 + lr;      // local row in LDS tile
    const int b   = b0 + row;          // global batch row
    float2 q0[8], q1[8];
#pragma unroll
    for (int i = 0; i < 8; ++i) {
      float xv = x[b * 8 + i];
      float cx = cosf(xv), sx = sinf(xv);                 // RZ(2x)
      float2 v0 = make_float2(cx * inv_s2, -sx * inv_s2);
      float2 v1 = make_float2(cx * inv_s2,  sx * inv_s2);
      float c1 = cosf(0.5f * xv), s1 = sinf(0.5f * xv);   // RY(x)
      float2 w0 = make_float2(c1 * v0.x - s1 * v1.x, c1 * v0.y - s1 * v1.y);
      float2 w1 = make_float2(s1 * v0.x + c1 * v1.x, s1 * v0.y + c1 * v1.y);
      float c2 = cosf(0.25f * xv), s2 = sinf(0.25f * xv); // RX(0.5x)
      q0[i] = make_float2(c2 * w0.x + s2 * w1.y, c2 * w0.y - s2 * w1.x);
      q1[i] = make_float2(c2 * w1.x + s2 * w0.y, c2 * w1.y - s2 * w0.x);
    }
    // amplitude index s = lane*8 + r: wires 0..4 <- lane bits 4..0,
    // wires 5..7 <- r bits 2..0.
    float2 pref = ((lane >> 4) & 1) ? q1[0] : q0[0];
#pragma unroll
    for (int i = 1; i < 5; ++i)
      pref = cmul(pref, ((lane >> (4 - i)) & 1) ? q1[i] : q0[i]);
#pragma unroll
    for (int r = 0; r < 8; r += 2) {
      float2 a5 = (r & 4) ? q1[5] : q0[5];
      float2 a6 = (r & 2) ? q1[6] : q0[6];
      float2 com  = cmul(cmul(pref, a5), a6);
      float2 amp0 = cmul(com, q0[7]);          // s even
      float2 amp1 = cmul(com, q1[7]);          // s odd
      v4f pk = {amp0.x, amp1.x, amp0.y, amp1.y};
      *(v4f*)(sP + row * SROW + (lane * 8 + r) * 2) = pk;
    }
  }
  __syncthreads();

  // ---- Phase 2: WMMA GEMM + fused reduction ----
  const int rowLocal = wv * 16 + lm;                 // A-fragment row
  const float* aRow = sP + rowLocal * SROW + half * 4;
  const float* bCol = W + (half * 256 + lm) * 4;     // + n0*64 + k'(row) below

  float partial[8];
#pragma unroll
  for (int j = 0; j < 8; ++j) partial[j] = 0.f;

  for (int n0 = 0; n0 < 16; ++n0) {
    v8f accRR = {0.f,0.f,0.f,0.f,0.f,0.f,0.f,0.f};
    v8f accII = {0.f,0.f,0.f,0.f,0.f,0.f,0.f,0.f};
    v8f accRI = {0.f,0.f,0.f,0.f,0.f,0.f,0.f,0.f};
    v8f accIR = {0.f,0.f,0.f,0.f,0.f,0.f,0.f,0.f};
    const float* bTile = bCol + n0 * 64;             // n = n0*16 + lm
#pragma unroll 4
    for (int k0 = 0; k0 < 256; k0 += 4) {
      // A: {re[kA], re[kA+1], im[kA], im[kA+1]}  (kA = k0 + 2*half)
      v4f Av = *(const v4f*)(aRow + k0 * 2);                 // ds_read_b128
      // B: same packing for column n, pair kA>>1
      v4f Bv = *(const v4f*)(bTile + (k0 >> 1) * 1024);      // global b128
      v2f ar = __builtin_shufflevector(Av, Av, 0, 1);
      v2f ai = __builtin_shufflevector(Av, Av, 2, 3);
      v2f br = __builtin_shufflevector(Bv, Bv, 0, 1);
      v2f bi = __builtin_shufflevector(Bv, Bv, 2, 3);
      accRR = WMMA_F32(ar, br, accRR);
      accII = WMMA_F32(ai, bi, accII);
      accRI = WMMA_F32(ar, bi, accRI);
      accIR = WMMA_F32(ai, br, accIR);
    }
    // sign(s) = parity(popcount(s & 0xF0)); s = n0*16 + lm -> depends on n0
    const float sgn = (__popc(n0) & 1) ? -1.f : 1.f;
#pragma unroll
    for (int j = 0; j < 8; ++j) {
      float orr = accRR[j] - accII[j];
      float oii = accRI[j] + accIR[j];
      partial[j] += sgn * (orr * orr + oii * oii);
    }
  }

  // Reduce over the 16 lanes of each half-wave (columns of the C tile).
#pragma unroll
  for (int m = 1; m < 16; m <<= 1) {
#pragma unroll
    for (int j = 0; j < 8; ++j)
      partial[j] += __shfl_xor(partial[j], m, 32);
  }
  if (lm == 0) {
#pragma unroll
    for (int j = 0; j < 8; ++j)
      out[b0 + wv * 16 + half * 8 + j] = partial[j];   // C row M = j + 8*half
  }
}

extern "C" void kernel_launch(void* const* d_in, const int* in_sizes, int n_in,
                              void* d_out, int out_size, void* d_ws,
                              size_t ws_size, hipStream_t stream) {
  const float* x = (const float*)d_in[0];   // (B, 8) f32
  const float* p = (const float*)d_in[1];   // (80,)  f32
  float* out = (float*)d_out;               // (B,)   f32

  float* W = (float*)d_ws;                  // 128*256*4 f32 = 512 KB

  const int Bn = in_sizes[0] / 8;           // 32768

  build_unitary<<<256, 256, 0, stream>>>(p, W);
  qnn_gemm_reduce<<<Bn / 64, 128, LDS_BYTES, stream>>>(x, W, out);
}